// MinusAttentionUnit_57552561766574
// MI455X (gfx1250) — compile-verified
//
#include <hip/hip_runtime.h>
#include <hip/hip_bf16.h>

typedef __attribute__((ext_vector_type(2))) float v2f;
typedef __attribute__((ext_vector_type(8))) float v8f;

#define B_DIM 4
#define LQ 512
#define LP 512
#define PQ 256
#define VDIM 256

// ---------------------------------------------------------------------------
// fast tanh / exp using gfx1250 TRANS ops (v_exp_f32, v_rcp_f32):
// tanh(x) = 1 - 2/(e^{2x}+1); saturates correctly at +/-inf.
// ---------------------------------------------------------------------------
__device__ __forceinline__ float fast_tanh(float x) {
    float e = __builtin_amdgcn_exp2f(x * 2.88539008177792681472f); // 2*log2(e)
    return 1.0f - 2.0f * __builtin_amdgcn_rcpf(e + 1.0f);
}
__device__ __forceinline__ float fast_exp(float x) {
    return __builtin_amdgcn_exp2f(x * 1.44269504088896340736f);
}

// ---------------------------------------------------------------------------
// f32 GEMM via V_WMMA_F32_16X16X4_F32.
//   C[b] (M x N, row-major) = A[b] (M x K, row-major) * B[b]
//   BT=true : Bmat is (N x K) row-major and used transposed (the W case);
//             B fragment is a contiguous float2 from a row of Bmat.
//   BT=false: Bmat is (K x N) row-major (the hq case).
// Block = 128 threads = 4 waves. Wave w computes rows [ (bx*4+w)*16 , +16 )
// and cols [ by*64 , +64 ) as 4 accumulators -> 4 WMMAs per k-step share one
// A fragment. Grid dims divide exactly -> EXEC all ones (WMMA requirement).
//
// VGPR layouts per CDNA5 ISA 7.12.2 (32-bit A 16x4 / C-D 16x16):
//   A frag:  lane L holds M=L%16, K = k0 + 2*(L/16) + {0,1} in v0,v1
//   B frag:  lane L holds N=L%16, K = k0 + 2*(L/16) + {0,1} in v0,v1
//   C tile:  vgpr r: lanes 0-15 -> (M=r, N=lane), lanes 16-31 -> (M=8+r)
// ---------------------------------------------------------------------------
template <bool BT>
__global__ void wmma_gemm_f32_kernel(const float* __restrict__ A,
                                     const float* __restrict__ Bmat,
                                     float* __restrict__ C,
                                     int M, int N, int K,
                                     long sA, long sB, long sC) {
    const int lane = threadIdx.x & 31;
    const int wave = threadIdx.x >> 5;
    const int b    = blockIdx.z;
    A    += (long)b * sA;
    Bmat += (long)b * sB;
    C    += (long)b * sC;

    const int m0    = (blockIdx.x * 4 + wave) * 16;
    const int n0    = blockIdx.y * 64;
    const int mrow  = m0 + (lane & 15);
    const int khalf = (lane >> 4) * 2;
    const int ncol  = lane & 15;

    v8f acc[4] = {v8f{}, v8f{}, v8f{}, v8f{}};

#pragma unroll 2
    for (int k0 = 0; k0 < K; k0 += 4) {
        const v2f av = *(const v2f*)(A + (long)mrow * K + (k0 + khalf));
#pragma unroll
        for (int t = 0; t < 4; ++t) {
            const int n = n0 + t * 16 + ncol;
            v2f bv;
            if (BT) {
                bv = *(const v2f*)(Bmat + (long)n * K + (k0 + khalf));
            } else {
                bv.x = Bmat[(long)(k0 + khalf) * N + n];
                bv.y = Bmat[(long)(k0 + khalf + 1) * N + n];
            }
            acc[t] = __builtin_amdgcn_wmma_f32_16x16x4_f32(
                false, av, false, bv, (short)0, acc[t], false, false);
        }
    }

    const int mbase = m0 + (lane >> 4) * 8;
#pragma unroll
    for (int t = 0; t < 4; ++t) {
        const int n = n0 + t * 16 + ncol;
#pragma unroll
        for (int r = 0; r < 8; ++r)
            C[(long)(mbase + r) * N + n] = acc[t][r];
    }
}

// ---------------------------------------------------------------------------
// Per-(b,p) row: scores s[q] = sum_v v[v]*tanh(Wq[b,q,v]-Wp[b,p,v]),
// softmax over q (512), write normalized attention row a[b,p,:].
// 256 threads/block (8 waves); thread t owns q = t and q = t+256.
// Wp row + v vector staged in LDS (float4), tree reductions in LDS.
// ---------------------------------------------------------------------------
__global__ void score_softmax_kernel(const float* __restrict__ Wq,   // (B*LQ, V)
                                     const float* __restrict__ Wp,   // (B*LP, V)
                                     const float* __restrict__ vvec, // (V)
                                     float* __restrict__ aout) {     // (B*LP, LQ)
    const int t  = threadIdx.x;      // 0..255
    const int bp = blockIdx.x;       // b*LP + p
    const int b  = bp >> 9;

    __shared__ float4 wp4[VDIM / 4];
    __shared__ float4 vv4[VDIM / 4];
    __shared__ float  red[256];

    if (t < VDIM / 4) {
        wp4[t] = ((const float4*)(Wp + (long)bp * VDIM))[t];
        vv4[t] = ((const float4*)vvec)[t];
    }
    __syncthreads();

    float s[2];
#pragma unroll
    for (int j = 0; j < 2; ++j) {
        const int q = t + j * 256;
        const float4* wq = (const float4*)(Wq + ((long)b * LQ + q) * VDIM);
        float acc = 0.0f;
#pragma unroll 4
        for (int c = 0; c < VDIM / 4; ++c) {
            const float4 w = wq[c];
            const float4 p = wp4[c];
            const float4 vv = vv4[c];
            acc += vv.x * fast_tanh(w.x - p.x);
            acc += vv.y * fast_tanh(w.y - p.y);
            acc += vv.z * fast_tanh(w.z - p.z);
            acc += vv.w * fast_tanh(w.w - p.w);
        }
        s[j] = acc;
    }

    // ---- max over 512 ----
    red[t] = fmaxf(s[0], s[1]);
    __syncthreads();
    for (int off = 128; off > 0; off >>= 1) {
        if (t < off) red[t] = fmaxf(red[t], red[t + off]);
        __syncthreads();
    }
    const float mx = red[0];
    __syncthreads();

    // ---- exp + sum ----
    const float e0 = fast_exp(s[0] - mx);
    const float e1 = fast_exp(s[1] - mx);
    red[t] = e0 + e1;
    __syncthreads();
    for (int off = 128; off > 0; off >>= 1) {
        if (t < off) red[t] += red[t + off];
        __syncthreads();
    }
    const float inv = __builtin_amdgcn_rcpf(red[0]);

    float* ao = aout + (long)bp * LQ;
    ao[t]       = e0 * inv;
    ao[t + 256] = e1 * inv;
}

// ---------------------------------------------------------------------------
// launch
// ---------------------------------------------------------------------------
extern "C" void kernel_launch(void* const* d_in, const int* in_sizes, int n_in,
                              void* d_out, int out_size, void* d_ws, size_t ws_size,
                              hipStream_t stream) {
    const float* hq = (const float*)d_in[0]; // (B, LQ, PQ)
    const float* hp = (const float*)d_in[1]; // (B, LP, PQ)
    const float* W  = (const float*)d_in[2]; // (V, PQ) row-major
    const float* v  = (const float*)d_in[3]; // (V)
    float* out = (float*)d_out;              // (B, LP, PQ)

    float* Wq = (float*)d_ws;                       // (B*LQ, V)  2 MB
    float* Wp = Wq + (size_t)B_DIM * LQ * VDIM;     // (B*LP, V)  2 MB
    float* a  = Wp + (size_t)B_DIM * LP * VDIM;     // (B*LP, LQ) 4 MB

    const dim3 blk(128);

    // Wq = hq @ W^T  (flatten batch: same W for all rows)
    wmma_gemm_f32_kernel<true><<<dim3((B_DIM * LQ) / 64, VDIM / 64, 1), blk, 0, stream>>>(
        hq, W, Wq, B_DIM * LQ, VDIM, PQ, 0, 0, 0);
    // Wp = hp @ W^T
    wmma_gemm_f32_kernel<true><<<dim3((B_DIM * LP) / 64, VDIM / 64, 1), blk, 0, stream>>>(
        hp, W, Wp, B_DIM * LP, VDIM, PQ, 0, 0, 0);

    // scores + softmax -> a
    score_softmax_kernel<<<B_DIM * LP, 256, 0, stream>>>(Wq, Wp, v, a);

    // q_att[b] = a[b] @ hq[b]   (M=LP, N=PQ, K=LQ, batched)
    wmma_gemm_f32_kernel<false><<<dim3(LP / 64, PQ / 64, B_DIM), blk, 0, stream>>>(
        a, hq, out, LP, PQ, LQ,
        (long)LP * LQ, (long)LQ * PQ, (long)LP * PQ);
}